// DisentangleGraph_77464030151239
// MI455X (gfx1250) — compile-verified
//
#include <hip/hip_runtime.h>
#include <hip/hip_bf16.h>

// ---------------------------------------------------------------------------
// MI455X (gfx1250) implementation.
// Roofline: dominant traffic = read H (328MB) + write H_out (328MB) ~= 656MB
//   -> ~28us at 23.3 TB/s. Both streams exceed the 192MB L2 and are touched
//   exactly once -> non-temporal loads/stores. The cosine-sim GEMM (82 MFLOP)
//   uses V_WMMA_F32_16X16X4_F32 (full f32: ranks of 20000 tightly packed
//   values must stay exact; fp16 would corrupt the top-6001 selection).
// ---------------------------------------------------------------------------

typedef __attribute__((ext_vector_type(2))) float v2f;
typedef __attribute__((ext_vector_type(4))) float vf4;
typedef __attribute__((ext_vector_type(8))) float v8f;

#define DIM 256
#define KFACT 8

// monotone float -> uint key (larger float => larger key)
__device__ __forceinline__ unsigned f2key(float x) {
    unsigned u = __float_as_uint(x);
    return (u & 0x80000000u) ? ~u : (u | 0x80000000u);
}

// ---------------- zero scratch accumulators -------------------------------
__global__ void k_zero(float* __restrict__ p, int n) {
    int i = blockIdx.x * 256 + threadIdx.x;
    if (i < n) p[i] = 0.0f;
}

// ---------------- B-pad (16x256, rows 8..15 zero) + inverse norms ---------
__global__ void k_prep(const float* __restrict__ int_emb,
                       float* __restrict__ bpad,
                       float* __restrict__ einv) {
    __shared__ float red[8];
    const int k = blockIdx.x;       // 0..15
    const int t = threadIdx.x;      // 256 threads
    float v = (k < KFACT) ? int_emb[(size_t)k * DIM + t] : 0.0f;
    bpad[(size_t)k * DIM + t] = v;
    if (k >= KFACT) return;
    float s = v * v;
    #pragma unroll
    for (int off = 16; off; off >>= 1) s += __shfl_xor(s, off, 32);
    if ((t & 31) == 0) red[t >> 5] = s;
    __syncthreads();
    if (t == 0) {
        float tot = 0.f;
        #pragma unroll
        for (int i = 0; i < 8; ++i) tot += red[i];
        einv[k] = 1.0f / fmaxf(sqrtf(tot), 1e-8f);
    }
}

// ---------------- WMMA cosine-sim -> sortable keys ------------------------
// One wave32 computes one 16-node x 16-factor tile via V_WMMA_F32_16X16X4_F32
// (B columns 8..15 are the zero-padded rows -> no divergence in the loop).
__global__ void k_sim(const float* __restrict__ hidden,
                      const float* __restrict__ bpad,
                      const float* __restrict__ einv,
                      unsigned* __restrict__ keys,
                      int n_nodes) {
    const int wavesPerBlock = blockDim.x >> 5;
    const int tile = blockIdx.x * wavesPerBlock + (threadIdx.x >> 5);
    if (tile * 16 >= n_nodes) return;              // wave-uniform exit
    const int lane = threadIdx.x & 31;
    const int m  = lane & 15;                      // A row / B-D column index
    const int kh = lane >> 4;                      // K-pair half (0 or 1)
    const int row = tile * 16 + m;

    const float* arow = hidden + (size_t)row * DIM + kh * 2;
    const float* brow = bpad   + (size_t)m   * DIM + kh * 2;

    v8f acc = {};
    float ssq0 = 0.0f, ssq1 = 0.0f;
    for (int kc = 0; kc < DIM; kc += 4) {
        v2f a, b;
        a.x = arow[kc]; a.y = arow[kc + 1];
        b.x = brow[kc]; b.y = brow[kc + 1];
        ssq0 = __builtin_fmaf(a.x, a.x, ssq0);
        ssq1 = __builtin_fmaf(a.y, a.y, ssq1);
        // v_wmma_f32_16x16x4_f32: D = A(16x4) * B(4x16) + C
        acc = __builtin_amdgcn_wmma_f32_16x16x4_f32(
            false, a, false, b, (short)0, acc, false, false);
    }
    float ssq = ssq0 + ssq1;
    // combine the two K-halves of the row sum-of-squares
    ssq += __shfl_xor(ssq, 16, 32);
    float rinv = 1.0f / fmaxf(sqrtf(ssq), 1e-8f);  // lane l holds row (l&15)

    if (m < KFACT) {
        float ei = einv[m];
        // D layout: lane l -> col n=l%16 ; VGPR r -> row r + 8*(l>>4)
        #pragma unroll
        for (int r = 0; r < 8; ++r) {
            int mr = r + 8 * kh;
            float rv = __shfl(rinv, mr, 32);
            float s = 10.0f * rv * ei * acc[r];
            keys[(size_t)(tile * 16 + mr) * KFACT + m] = f2key(s);
        }
    }
}

// ---------------- exact top-(sel+1) threshold: 4-pass LDS radix select ----
__global__ void k_select(const unsigned* __restrict__ keys, int n_nodes,
                         unsigned sel_plus1,
                         unsigned* __restrict__ thrKey,
                         unsigned* __restrict__ thrR) {
    __shared__ unsigned hist[256];
    __shared__ unsigned s_prefix, s_k;
    const int f = blockIdx.x;
    const int t = threadIdx.x;                      // 256 threads
    if (t == 0) { s_prefix = 0u; s_k = sel_plus1; }
    __syncthreads();
    for (int p = 3; p >= 0; --p) {
        hist[t] = 0u;
        __syncthreads();
        const unsigned prefix = s_prefix;
        const unsigned maskAbove = (p == 3) ? 0u : (0xFFFFFFFFu << ((p + 1) * 8));
        const int shift = p * 8;
        for (int n = t; n < n_nodes; n += 256) {
            unsigned key = keys[(size_t)n * KFACT + f];
            if (((key ^ prefix) & maskAbove) == 0u)
                atomicAdd(&hist[(key >> shift) & 255u], 1u);
        }
        __syncthreads();
        if (t == 0) {
            unsigned k = s_k, pref = s_prefix;
            for (int b = 255; b >= 0; --b) {
                unsigned c = hist[b];
                if (k <= c) { pref |= ((unsigned)b << shift); break; }
                k -= c;
            }
            s_k = k; s_prefix = pref;
        }
        __syncthreads();
    }
    if (t == 0) { thrKey[f] = s_prefix; thrR[f] = s_k; }
}

// ---------------- mark selected nodes (stable tie-break by index) ---------
__global__ void k_mark(const unsigned* __restrict__ keys, int n_nodes,
                       const unsigned* __restrict__ thrKey,
                       const unsigned* __restrict__ thrR,
                       unsigned char* __restrict__ selbytes) {
    __shared__ unsigned s_wcnt[32];
    __shared__ unsigned s_run;
    const int f = blockIdx.x;
    const unsigned T = thrKey[f];
    const unsigned R = thrR[f];                     // # equals to accept, index order
    const int tid = threadIdx.x;                    // 1024 threads = 32 waves
    const int lane = tid & 31, wid = tid >> 5;
    if (tid == 0) s_run = 0u;
    __syncthreads();
    for (int base = 0; base < n_nodes; base += 1024) {
        int n = base + tid;
        bool valid = n < n_nodes;
        unsigned key = valid ? keys[(size_t)n * KFACT + f] : 0u;
        bool eq = valid && (key == T);
        bool gt = valid && (key > T);
        unsigned bal = (unsigned)__ballot(eq);      // wave32: low 32 bits
        unsigned pref = __popc(bal & ((1u << lane) - 1u));
        if (lane == 0) s_wcnt[wid] = __popc(bal);
        __syncthreads();
        unsigned woff = 0;
        for (int w = 0; w < wid; ++w) woff += s_wcnt[w];
        unsigned eqidx = s_run + woff + pref;
        if (valid)
            selbytes[(size_t)f * n_nodes + n] = (gt || (eq && eqidx < R)) ? 1 : 0;
        __syncthreads();
        if (tid == 0) {
            unsigned tot = 0;
            for (int w = 0; w < 32; ++w) tot += s_wcnt[w];
            s_run += tot;
        }
        __syncthreads();
    }
}

// ---------------- the big fused streaming pass over H ---------------------
// Per row: NT-load H row once (16 floats/thread in regs), block-reduce degV,
// NT-store [int_H | H] to H_out, accumulate per-column cnt / sum(mask*degV)
// in registers (thread-private columns), flush via global f32 atomics at end.
#define ROWS_PER_BLOCK 50
__global__ void k_fuse(const float* __restrict__ H,
                       const unsigned char* __restrict__ selbytes,
                       float* __restrict__ Hout,
                       float* __restrict__ degV_out,
                       float* __restrict__ colsum,
                       float* __restrict__ colcnt,
                       int n_nodes, int n_edge) {
    __shared__ float s_red[8];
    __shared__ float s_degv;
    const int t = threadIdx.x;                      // 256
    const int lane = t & 31, wid = t >> 5;
    const int row0 = blockIdx.x * ROWS_PER_BLOCK;

    float cacc[16], sacc[16];
    #pragma unroll
    for (int j = 0; j < 16; ++j) { cacc[j] = 0.f; sacc[j] = 0.f; }
    float cint = 0.f, sint = 0.f;

    for (int r = 0; r < ROWS_PER_BLOCK; ++r) {
        const int n = row0 + r;
        if (n >= n_nodes) break;                    // block-uniform
        const vf4* hrow = (const vf4*)(H + (size_t)n * n_edge);

        vf4 v[4];
        float partial = 0.f;
        #pragma unroll
        for (int i = 0; i < 4; ++i) {
            v[i] = __builtin_nontemporal_load(&hrow[i * 256 + t]);
            partial += v[i].x + v[i].y + v[i].z + v[i].w;
        }
        float intval = 0.f;
        unsigned char selb = 0;
        if (t < KFACT) {
            selb = selbytes[(size_t)t * n_nodes + n];
            intval = selb ? 2.0f : 0.0f;
            partial += intval;
        }
        // block reduction -> degV (raw row sum of H_out)
        #pragma unroll
        for (int off = 16; off; off >>= 1) partial += __shfl_xor(partial, off, 32);
        if (lane == 0) s_red[wid] = partial;
        __syncthreads();
        if (t == 0) {
            float d = 0.f;
            #pragma unroll
            for (int i = 0; i < 8; ++i) d += s_red[i];
            s_degv = d;
            degV_out[n] = (d > 0.f) ? rsqrtf(d) : 1.0f;   // inf -> 1.0 rule
        }
        __syncthreads();
        const float degv = s_degv;

        // write H_out row = [int_H(8) | H(4096)]; +8 floats keeps 16B align
        float* obase = Hout + (size_t)n * (n_edge + KFACT);
        if (t < KFACT) obase[t] = intval;
        vf4* orow = (vf4*)(obase + KFACT);
        #pragma unroll
        for (int i = 0; i < 4; ++i)
            __builtin_nontemporal_store(v[i], &orow[i * 256 + t]);

        // column accumulation in registers (thread-private columns)
        #pragma unroll
        for (int i = 0; i < 4; ++i) {
            float mx = (v[i].x != 0.f) ? 1.f : 0.f;
            float my = (v[i].y != 0.f) ? 1.f : 0.f;
            float mz = (v[i].z != 0.f) ? 1.f : 0.f;
            float mw = (v[i].w != 0.f) ? 1.f : 0.f;
            cacc[i*4+0] += mx; sacc[i*4+0] = __builtin_fmaf(mx, degv, sacc[i*4+0]);
            cacc[i*4+1] += my; sacc[i*4+1] = __builtin_fmaf(my, degv, sacc[i*4+1]);
            cacc[i*4+2] += mz; sacc[i*4+2] = __builtin_fmaf(mz, degv, sacc[i*4+2]);
            cacc[i*4+3] += mw; sacc[i*4+3] = __builtin_fmaf(mw, degv, sacc[i*4+3]);
        }
        if (t < KFACT) {
            float m = selb ? 1.f : 0.f;
            cint += m; sint = __builtin_fmaf(m, degv, sint);
        }
        __syncthreads();                            // s_red/s_degv reuse
    }
    // flush per-thread column partials
    #pragma unroll
    for (int i = 0; i < 4; ++i) {
        #pragma unroll
        for (int j = 0; j < 4; ++j) {
            int col = KFACT + 4 * (i * 256 + t) + j;
            atomicAdd(&colcnt[col], cacc[i * 4 + j]);
            atomicAdd(&colsum[col], sacc[i * 4 + j]);
        }
    }
    if (t < KFACT) {
        atomicAdd(&colcnt[t], cint);
        atomicAdd(&colsum[t], sint);
    }
}

// ---------------- degE finalize -------------------------------------------
__global__ void k_finalize(const float* __restrict__ colsum,
                           const float* __restrict__ colcnt,
                           float* __restrict__ degE_out, int ncols) {
    int e = blockIdx.x * 256 + threadIdx.x;
    if (e < ncols) {
        float val = colsum[e] / fmaxf(colcnt[e], 1.0f);
        degE_out[e] = rsqrtf(val);                  // 0 -> +inf, matches x**-0.5
    }
}

// ---------------------------------------------------------------------------
extern "C" void kernel_launch(void* const* d_in, const int* in_sizes, int n_in,
                              void* d_out, int out_size, void* d_ws, size_t ws_size,
                              hipStream_t stream) {
    const float* hidden  = (const float*)d_in[0];   // (N, 256)
    const float* H       = (const float*)d_in[1];   // (N, E)
    const float* int_emb = (const float*)d_in[2];   // (8, 256)

    const int N = in_sizes[0] / DIM;                // 20000
    const int E = in_sizes[1] / N;                  // 4096
    const int NC = KFACT + E;                       // 4104
    const unsigned sel_plus1 = (unsigned)((long long)floor(0.3 * (double)N)) + 1u;

    // workspace layout
    char* ws = (char*)d_ws;
    unsigned*       w_keys   = (unsigned*)(ws + 0);               // N*8 u32
    float*          w_einv   = (float*)   (ws + 640000);          // 8 f32
    unsigned*       w_thrKey = (unsigned*)(ws + 640064);          // 8 u32
    unsigned*       w_thrR   = (unsigned*)(ws + 640128);          // 8 u32
    float*          w_colsum = (float*)   (ws + 640192);          // NC f32
    float*          w_colcnt = (float*)   (ws + 656640);          // NC f32
    unsigned char*  w_sel    = (unsigned char*)(ws + 673280);     // 8*N u8
    float*          w_bpad   = (float*)   (ws + 833536);          // 16*256 f32

    float* Hout     = (float*)d_out;
    float* degV_out = Hout + (size_t)N * NC;
    float* degE_out = degV_out + N;

    // 1) zero column accumulators (must happen every call; ws is dirty)
    k_zero<<<(NC + 255) / 256, 256, 0, stream>>>(w_colsum, NC);
    k_zero<<<(NC + 255) / 256, 256, 0, stream>>>(w_colcnt, NC);

    // 2) zero-padded B (16x256) + intent-embedding inverse norms
    k_prep<<<16, DIM, 0, stream>>>(int_emb, w_bpad, w_einv);

    // 3) WMMA f32 cosine-sim -> sortable keys (1 wave per 16-node tile)
    {
        int tiles = (N + 15) / 16;                  // 1250
        int wavesPerBlock = 8;                      // 256 threads
        int grid = (tiles + wavesPerBlock - 1) / wavesPerBlock;
        k_sim<<<grid, 32 * wavesPerBlock, 0, stream>>>(hidden, w_bpad, w_einv,
                                                       w_keys, N);
    }

    // 4) per-factor exact radix select of the (sel+1)-th largest key
    k_select<<<KFACT, 256, 0, stream>>>(w_keys, N, sel_plus1, w_thrKey, w_thrR);

    // 5) membership marking with stable index-order tie-break
    k_mark<<<KFACT, 1024, 0, stream>>>(w_keys, N, w_thrKey, w_thrR, w_sel);

    // 6) fused streaming pass: H -> H_out, degV, column partial sums
    {
        int grid = (N + ROWS_PER_BLOCK - 1) / ROWS_PER_BLOCK;   // 400
        k_fuse<<<grid, 256, 0, stream>>>(H, w_sel, Hout, degV_out,
                                         w_colsum, w_colcnt, N, E);
    }

    // 7) degE finalize
    k_finalize<<<(NC + 255) / 256, 256, 0, stream>>>(w_colsum, w_colcnt,
                                                     degE_out, NC);
}